// GATPair_38800734552870
// MI455X (gfx1250) — compile-verified
//
#include <hip/hip_runtime.h>
#include <hip/hip_bf16.h>
#include <math.h>

typedef __attribute__((ext_vector_type(2))) float v2f;
typedef __attribute__((ext_vector_type(8))) float v8f;

#define N_NODES 100000
#define N_EDGES 1600000
#define D_IN    128
#define HID     64
#define HEADS   2
#define OUTD    2
#define BATCH   8192
#define NEG_SLOPE 0.2f

// ---------------------------------------------------------------------------
// Strip-mined WMMA f32 GEMM: each wave computes a 16 x (NTILES*16) row strip
// of C = A[M,K] @ B[K, NTILES*16].  K, NTILES compile-time -> all address
// math folds into immediate offsets; A fragment reused across NTILES WMMAs.
// Optionally fuses the GAT attention-score reduction (s_src/s_dst) into the
// epilogue while the h-strip is still in registers (requires NTILES==8,
// i.e. 128 output channels = 2 heads x 64).
// ---------------------------------------------------------------------------
template <int K, int NTILES, bool FUSE_SCORES>
__global__ void gemm_wmma_strip(const float* __restrict__ A,
                                const float* __restrict__ Bm,
                                float* __restrict__ C, int M,
                                const float* __restrict__ asrc,
                                const float* __restrict__ adst,
                                float* __restrict__ ssrc,
                                float* __restrict__ sdst) {
    constexpr int NCOL = NTILES * 16;
    int wave = (int)((blockIdx.x * blockDim.x + threadIdx.x) >> 5);
    int rows = M >> 4;
    if (wave >= rows) return;                  // wave-uniform: EXEC stays all-1s
    int lane = (int)(threadIdx.x & 31);
    int half = lane >> 4;                      // 0: K pair {k,k+1}, 1: {k+2,k+3}
    int l15  = lane & 15;

    const float* arow = A  + (size_t)(wave * 16 + l15) * K + half * 2;
    const float* bcol = Bm + (size_t)half * 2 * NCOL + l15;

    v8f acc[NTILES];
#pragma unroll
    for (int t = 0; t < NTILES; ++t)
        acc[t] = (v8f){0.f,0.f,0.f,0.f,0.f,0.f,0.f,0.f};

#pragma unroll 4
    for (int k = 0; k < K; k += 4) {
        v2f av; av.x = arow[k];
                av.y = arow[k + 1];
        const float* bk = bcol + (size_t)k * NCOL;
#pragma unroll
        for (int t = 0; t < NTILES; ++t) {
            v2f bv; bv.x = bk[t * 16];
                    bv.y = bk[t * 16 + NCOL];
            acc[t] = __builtin_amdgcn_wmma_f32_16x16x4_f32(
                        false, av, false, bv, (short)0, acc[t], false, false);
        }
    }

    // D layout: VGPR r -> M = r + 8*half, N = t*16 + (lane&15)
    float* crow = C + (size_t)(wave * 16 + half * 8) * NCOL + l15;
#pragma unroll
    for (int t = 0; t < NTILES; ++t)
#pragma unroll
        for (int r = 0; r < 8; ++r)
            crow[(size_t)r * NCOL + t * 16] = acc[t][r];

    if (FUSE_SCORES) {
        // channel for (t, lane) is c = t*16 + l15; head = t>>2 (c/64).
        float av_[NTILES], dv_[NTILES];
#pragma unroll
        for (int t = 0; t < NTILES; ++t) {
            av_[t] = asrc[t * 16 + l15];
            dv_[t] = adst[t * 16 + l15];
        }
#pragma unroll
        for (int r = 0; r < 8; ++r) {
            float s0 = 0.f, s1 = 0.f, d0 = 0.f, d1 = 0.f;
#pragma unroll
            for (int t = 0; t < NTILES; ++t) {
                float hv = acc[t][r];
                if (t < NTILES / 2) { s0 += hv * av_[t]; d0 += hv * dv_[t]; }
                else                { s1 += hv * av_[t]; d1 += hv * dv_[t]; }
            }
            // reduce over the 16 lanes of this half (xor 1,2,4,8 stay in-half)
#pragma unroll
            for (int off = 1; off < 16; off <<= 1) {
                s0 += __shfl_xor(s0, off); s1 += __shfl_xor(s1, off);
                d0 += __shfl_xor(d0, off); d1 += __shfl_xor(d1, off);
            }
            if (l15 == 0) {
                int m = wave * 16 + half * 8 + r;
                ssrc[m * 2 + 0] = s0; ssrc[m * 2 + 1] = s1;
                sdst[m * 2 + 0] = d0; sdst[m * 2 + 1] = d1;
            }
        }
    }
}

__global__ void gat_init_k(float* __restrict__ acc, float* __restrict__ mbuf,
                           float* __restrict__ zbuf, int n) {
    int i = (int)(blockIdx.x * blockDim.x + threadIdx.x);
    if (i < n * (HEADS * HID)) acc[i] = 0.f;
    if (i < n * HEADS) { mbuf[i] = -INFINITY; zbuf[i] = 0.f; }
}

__device__ __forceinline__ float lrelu(float x) {
    return x > 0.f ? x : NEG_SLOPE * x;
}

// Edge index i in [0, E+N): i<E -> real edge, else self loop (src=dst=i-E).
__global__ void edge_max_k(const int* __restrict__ src, const int* __restrict__ dst,
                           const float* __restrict__ ssrc, const float* __restrict__ sdst,
                           float* __restrict__ mbuf, int e, int n) {
    int i = (int)(blockIdx.x * blockDim.x + threadIdx.x);
    if (i >= e + n) return;
    int s, d;
    if (i < e) { s = src[i]; d = dst[i]; } else { s = i - e; d = s; }
#pragma unroll
    for (int hd = 0; hd < HEADS; ++hd) {
        float ev = lrelu(ssrc[s * 2 + hd] + sdst[d * 2 + hd]);
        atomicMax(&mbuf[d * 2 + hd], ev);     // global_atomic_max_num_f32
    }
}

__global__ void edge_sum_k(const int* __restrict__ src, const int* __restrict__ dst,
                           const float* __restrict__ ssrc, const float* __restrict__ sdst,
                           const float* __restrict__ mbuf, float* __restrict__ zbuf,
                           int e, int n) {
    int i = (int)(blockIdx.x * blockDim.x + threadIdx.x);
    if (i >= e + n) return;
    int s, d;
    if (i < e) { s = src[i]; d = dst[i]; } else { s = i - e; d = s; }
#pragma unroll
    for (int hd = 0; hd < HEADS; ++hd) {
        float ev = lrelu(ssrc[s * 2 + hd] + sdst[d * 2 + hd]);
        atomicAdd(&zbuf[d * 2 + hd], expf(ev - mbuf[d * 2 + hd]));
    }
}

// One wave per edge; lane handles 4 channels (float4 of h[src]); f32 atomics
// into acc[dst].  Working set (h + acc ~ 102 MB) is L2-resident on MI455X.
__global__ void edge_agg_k(const int* __restrict__ src, const int* __restrict__ dst,
                           const float* __restrict__ ssrc, const float* __restrict__ sdst,
                           const float* __restrict__ mbuf, const float* __restrict__ zbuf,
                           const float* __restrict__ h, float* __restrict__ acc,
                           int e, int n) {
    int t = (int)(blockIdx.x * blockDim.x + threadIdx.x);
    int i = t >> 5;
    if (i >= e + n) return;
    int lane = t & 31;
    int s, d;
    if (i < e) { s = src[i]; d = dst[i]; } else { s = i - e; d = s; }
    int hd = lane >> 4;                        // channels lane*4.. -> head lane/16
    float ev    = lrelu(ssrc[s * 2 + hd] + sdst[d * 2 + hd]);
    float alpha = expf(ev - mbuf[d * 2 + hd]) / zbuf[d * 2 + hd];
    const float4 hv = *(const float4*)(h + (size_t)s * (HEADS * HID) + lane * 4);
    float* ap = acc + (size_t)d * (HEADS * HID) + lane * 4;
    atomicAdd(ap + 0, hv.x * alpha);
    atomicAdd(ap + 1, hv.y * alpha);
    atomicAdd(ap + 2, hv.z * alpha);
    atomicAdd(ap + 3, hv.w * alpha);
}

// mean over heads + bias + relu  ->  node features [n, HID]
__global__ void gat_finalize_k(const float* __restrict__ acc,
                               const float* __restrict__ bias,
                               float* __restrict__ outf, int n) {
    int i = (int)(blockIdx.x * blockDim.x + threadIdx.x);
    if (i >= n * HID) return;
    int node = i >> 6, c = i & 63;
    float v = 0.5f * (acc[(size_t)node * 128 + c] + acc[(size_t)node * 128 + 64 + c])
              + bias[c];
    outf[i] = fmaxf(v, 0.f);
}

// merged[b, 0:64] = featL[labels_l[b]] ; merged[b, 64:128] = featR[labels_r[b]]
__global__ void gather_merge_k(const float* __restrict__ featL,
                               const float* __restrict__ featR,
                               const int* __restrict__ ll, const int* __restrict__ lr,
                               float* __restrict__ merged, int bn) {
    int i = (int)(blockIdx.x * blockDim.x + threadIdx.x);
    if (i >= bn * 128) return;
    int b = i >> 7, c = i & 127;
    merged[i] = (c < 64) ? featL[(size_t)ll[b] * 64 + c]
                         : featR[(size_t)lr[b] * 64 + (c - 64)];
}

__global__ void bias_relu_k(float* __restrict__ t, const float* __restrict__ bias,
                            int rows, int cols) {
    int i = (int)(blockIdx.x * blockDim.x + threadIdx.x);
    if (i >= rows * cols) return;
    t[i] = fmaxf(t[i] + bias[i % cols], 0.f);
}

__global__ void fc2_k(const float* __restrict__ t, const float* __restrict__ w,
                      const float* __restrict__ b, float* __restrict__ out, int bn) {
    int i = (int)(blockIdx.x * blockDim.x + threadIdx.x);
    if (i >= bn * OUTD) return;
    int row = i >> 1, o = i & 1;
    float s = b[o];
    const float* tr = t + (size_t)row * HID;
#pragma unroll 8
    for (int c = 0; c < HID; ++c) s += tr[c] * w[c * OUTD + o];
    out[i] = s;
}

// ---------------------------------------------------------------------------
// Host orchestration
// ---------------------------------------------------------------------------
static void run_gat_layer(const float* x, int fin,
                          const int* src, const int* dst,
                          const float* W, const float* asrc, const float* adst,
                          const float* bias,
                          float* hbuf, float* accb, float* ssrc, float* sdst,
                          float* mbuf, float* zbuf, float* outfeat,
                          hipStream_t stream) {
    const int n = N_NODES, e = N_EDGES;
    // h = x @ W  ([n,fin] x [fin,128]) with fused attention-score epilogue
    {
        int waves = n / 16;                    // one wave per 16-row strip
        int blocks = (waves + 7) / 8;
        if (fin == 128)
            gemm_wmma_strip<128, 8, true><<<blocks, 256, 0, stream>>>(
                x, W, hbuf, n, asrc, adst, ssrc, sdst);
        else
            gemm_wmma_strip<64, 8, true><<<blocks, 256, 0, stream>>>(
                x, W, hbuf, n, asrc, adst, ssrc, sdst);
    }
    gat_init_k<<<(n * 128 + 255) / 256, 256, 0, stream>>>(accb, mbuf, zbuf, n);
    int tot = e + n;
    edge_max_k<<<(tot + 255) / 256, 256, 0, stream>>>(src, dst, ssrc, sdst, mbuf, e, n);
    edge_sum_k<<<(tot + 255) / 256, 256, 0, stream>>>(src, dst, ssrc, sdst, mbuf, zbuf, e, n);
    edge_agg_k<<<((size_t)tot * 32 + 255) / 256, 256, 0, stream>>>(
        src, dst, ssrc, sdst, mbuf, zbuf, hbuf, accb, e, n);
    gat_finalize_k<<<(n * HID + 255) / 256, 256, 0, stream>>>(accb, bias, outfeat, n);
}

extern "C" void kernel_launch(void* const* d_in, const int* in_sizes, int n_in,
                              void* d_out, int out_size, void* d_ws, size_t ws_size,
                              hipStream_t stream) {
    (void)in_sizes; (void)n_in; (void)out_size; (void)ws_size;
    const float* x_l = (const float*)d_in[0];
    const float* x_r = (const float*)d_in[1];
    const int* ei_l  = (const int*)d_in[2];
    const int* ei_r  = (const int*)d_in[3];
    const int* lbl_l = (const int*)d_in[4];
    const int* lbl_r = (const int*)d_in[5];
    const float *w1l = (const float*)d_in[6],  *as1l = (const float*)d_in[7],
                *ad1l = (const float*)d_in[8], *b1l = (const float*)d_in[9];
    const float *w2l = (const float*)d_in[10], *as2l = (const float*)d_in[11],
                *ad2l = (const float*)d_in[12], *b2l = (const float*)d_in[13];
    const float *w1r = (const float*)d_in[14], *as1r = (const float*)d_in[15],
                *ad1r = (const float*)d_in[16], *b1r = (const float*)d_in[17];
    const float *w2r = (const float*)d_in[18], *as2r = (const float*)d_in[19],
                *ad2r = (const float*)d_in[20], *b2r = (const float*)d_in[21];
    const float *fc1w = (const float*)d_in[22], *fc1b = (const float*)d_in[23];
    const float *fc2w = (const float*)d_in[24], *fc2b = (const float*)d_in[25];
    float* out = (float*)d_out;

    const int* src_l = ei_l;            const int* dst_l = ei_l + N_EDGES;
    const int* src_r = ei_r;            const int* dst_r = ei_r + N_EDGES;

    // workspace carve-up (floats)
    float* ws    = (float*)d_ws;
    float* hbuf  = ws;                                   // N*128
    float* accb  = hbuf  + (size_t)N_NODES * 128;        // N*128
    float* ssrc  = accb  + (size_t)N_NODES * 128;        // N*2
    float* sdst  = ssrc  + (size_t)N_NODES * 2;          // N*2
    float* mbuf  = sdst  + (size_t)N_NODES * 2;          // N*2
    float* zbuf  = mbuf  + (size_t)N_NODES * 2;          // N*2
    float* featL = zbuf  + (size_t)N_NODES * 2;          // N*64
    float* featR = featL + (size_t)N_NODES * 64;         // N*64
    float* merged= featR + (size_t)N_NODES * 64;         // B*128
    float* t1    = merged+ (size_t)BATCH * 128;          // B*64

    // left tower
    run_gat_layer(x_l, 128, src_l, dst_l, w1l, as1l, ad1l, b1l,
                  hbuf, accb, ssrc, sdst, mbuf, zbuf, featL, stream);
    run_gat_layer(featL, 64, src_l, dst_l, w2l, as2l, ad2l, b2l,
                  hbuf, accb, ssrc, sdst, mbuf, zbuf, featL, stream);
    // right tower
    run_gat_layer(x_r, 128, src_r, dst_r, w1r, as1r, ad1r, b1r,
                  hbuf, accb, ssrc, sdst, mbuf, zbuf, featR, stream);
    run_gat_layer(featR, 64, src_r, dst_r, w2r, as2r, ad2r, b2r,
                  hbuf, accb, ssrc, sdst, mbuf, zbuf, featR, stream);

    // merge + MLP head
    gather_merge_k<<<(BATCH * 128 + 255) / 256, 256, 0, stream>>>(
        featL, featR, lbl_l, lbl_r, merged, BATCH);
    {
        int waves = BATCH / 16;                          // 512 row strips
        gemm_wmma_strip<128, 4, false><<<(waves + 7) / 8, 256, 0, stream>>>(
            merged, fc1w, t1, BATCH, nullptr, nullptr, nullptr, nullptr);
    }
    bias_relu_k<<<(BATCH * HID + 255) / 256, 256, 0, stream>>>(t1, fc1b, BATCH, HID);
    fc2_k<<<(BATCH * OUTD + 255) / 256, 256, 0, stream>>>(t1, fc2w, fc2b, out, BATCH);
}